// AttnDecoderRNN_58153857188588
// MI455X (gfx1250) — compile-verified
//
#include <hip/hip_runtime.h>
#include <math.h>

#define HSZ 1024
#define VSZ 50257
#define LSZ 40

typedef float v2f __attribute__((ext_vector_type(2)));
typedef float v4f __attribute__((ext_vector_type(4)));
typedef float v8f __attribute__((ext_vector_type(8)));

// ---- CDNA5 async copy: global -> LDS, 16B per active lane (ASYNCcnt) ----
__device__ __forceinline__ void async_copy_b128(unsigned lds_addr, const void* gaddr) {
    asm volatile("global_load_async_to_lds_b128 %0, %1, off"
                 :: "v"(lds_addr), "v"(gaddr) : "memory");
}
__device__ __forceinline__ void wait_async0() {
#if defined(__has_builtin)
#if __has_builtin(__builtin_amdgcn_s_wait_asynccnt)
    __builtin_amdgcn_s_wait_asynccnt(0);
#else
    asm volatile("s_wait_asynccnt 0" ::: "memory");
#endif
#else
    asm volatile("s_wait_asynccnt 0" ::: "memory");
#endif
}

// ---------------- Kernel 1: embedding + attention + comb_in ----------------
__global__ void attn_kernel(const int* __restrict__ tok_p,
                            const float* __restrict__ hidden,
                            const float* __restrict__ enc,
                            const float* __restrict__ emb,
                            const float* __restrict__ W_attn,
                            const float* __restrict__ b_attn,
                            float* __restrict__ d_out,
                            float* __restrict__ comb_in)
{
    __shared__ __align__(16) float attn_in[2 * HSZ];
    __shared__ float sc[LSZ];
    __shared__ float aw[LSZ];
    const int tid = threadIdx.x;
    const int tok = tok_p[0];
    for (int i = tid; i < HSZ; i += 256) {
        attn_in[i]       = emb[(size_t)tok * HSZ + i];
        attn_in[HSZ + i] = hidden[i];
    }
    __syncthreads();
    const int lane = tid & 31, wv = tid >> 5;
    for (int l = wv; l < LSZ; l += 8) {
        float p = 0.f;
        const float* wr = W_attn + (size_t)l * (2 * HSZ);
        for (int i = lane; i < 2 * HSZ; i += 32) p += wr[i] * attn_in[i];
        for (int off = 16; off > 0; off >>= 1) p += __shfl_xor(p, off, 32);
        if (lane == 0) sc[l] = p + b_attn[l];
    }
    __syncthreads();
    if (tid == 0) {
        float m = -INFINITY;
        for (int l = 0; l < LSZ; ++l) m = fmaxf(m, sc[l]);
        float s = 0.f;
        for (int l = 0; l < LSZ; ++l) { float e = __expf(sc[l] - m); aw[l] = e; s += e; }
        float inv = 1.f / s;
        for (int l = 0; l < LSZ; ++l) { aw[l] *= inv; d_out[VSZ + HSZ + l] = aw[l]; }
    }
    __syncthreads();
    for (int j = tid; j < HSZ; j += 256) {
        float acc = 0.f;
        for (int l = 0; l < LSZ; ++l) acc += aw[l] * enc[(size_t)l * HSZ + j];
        comb_in[j]       = attn_in[j];
        comb_in[HSZ + j] = acc;
    }
}

// ---------------- Generic GEMV via V_WMMA_F32_16X16X4_F32 ----------------
// y[m] = act(dot(W[m,:], x) + b[m]).  One wave per 16-row tile.
// K-permuted layout: within each 8-float chunk, lanes 0-15 own floats 0..3
// (K=0,1 then K=2,3 of two WMMAs), lanes 16-31 own floats 4..7.  One b128
// global load (non-temporal) + one b128 LDS read feed two WMMAs.
// D: lane 0 holds M=0..7 in c[0..7]; lane 16 holds M=8..15 (all N equal).
__global__ void gemv_wmma(const float* __restrict__ W,
                          const float* __restrict__ x,
                          const float* __restrict__ b,
                          float* __restrict__ y,
                          int M, int K, int relu)
{
    __shared__ __align__(16) float xs[2 * HSZ];
    const int tid = threadIdx.x;

    // stage x into LDS via CDNA5 async copies (K is a multiple of 1024)
    for (int i = tid * 4; i < K; i += 256 * 4) {
        unsigned lds = (unsigned)(unsigned long long)(&xs[i]);
        async_copy_b128(lds, (const void*)(x + i));
    }
    wait_async0();
    __syncthreads();

    const int lane = tid & 31;
    const int wv   = tid >> 5;
    const int half = lane >> 4;        // 0: chunk floats 0..3,  1: floats 4..7
    const int mrow = lane & 15;
    const int base = (blockIdx.x * 8 + wv) * 16;
    int r = base + mrow;
    if (r >= M) r = M - 1;             // clamp loads; writes guarded below
    const float* wp = W + (size_t)r * K;
    const int koff = 4 * half;

    v8f c0 = {};
    v8f c1 = {};
    for (int k0 = 0; k0 < K; k0 += 16) {
        __builtin_prefetch(wp + k0 + 1024, 0, 0);   // stream weight row ahead
        v4f a0 = __builtin_nontemporal_load((const v4f*)(wp + k0 + koff));
        v4f b0 = *(const v4f*)(xs + k0 + koff);
        v4f a1 = __builtin_nontemporal_load((const v4f*)(wp + k0 + 8 + koff));
        v4f b1 = *(const v4f*)(xs + k0 + 8 + koff);
        c0 = __builtin_amdgcn_wmma_f32_16x16x4_f32(false, a0.xy, false, b0.xy,
                                                   (short)0, c0, false, false);
        c1 = __builtin_amdgcn_wmma_f32_16x16x4_f32(false, a0.zw, false, b0.zw,
                                                   (short)0, c1, false, false);
        c0 = __builtin_amdgcn_wmma_f32_16x16x4_f32(false, a1.xy, false, b1.xy,
                                                   (short)0, c0, false, false);
        c1 = __builtin_amdgcn_wmma_f32_16x16x4_f32(false, a1.zw, false, b1.zw,
                                                   (short)0, c1, false, false);
    }
    c0 += c1;   // all D columns identical (B columns broadcast)

    if (lane == 0 || lane == 16) {
        const int mbase = base + 8 * half;
        for (int j = 0; j < 8; ++j) {
            const int m = mbase + j;
            if (m < M) {
                float v = c0[j] + b[m];
                if (relu) v = fmaxf(v, 0.f);
                y[m] = v;
            }
        }
    }
}

// ---------------- GRU gate combine ----------------
__global__ void gru_kernel(const float* __restrict__ gi,
                           const float* __restrict__ gh,
                           const float* __restrict__ hidden,
                           float* __restrict__ d_out,
                           float* __restrict__ hnew_ws)
{
    const int i = blockIdx.x * 256 + threadIdx.x;
    if (i < HSZ) {
        float r = 1.f / (1.f + __expf(-(gi[i]           + gh[i])));
        float z = 1.f / (1.f + __expf(-(gi[i + HSZ]     + gh[i + HSZ])));
        float n = tanhf(gi[i + 2 * HSZ] + r * gh[i + 2 * HSZ]);
        float hv = hidden[i];
        float hn = (1.f - z) * n + z * hv;
        d_out[VSZ + i] = hn;
        hnew_ws[i] = hn;
    }
}

// ---------------- log-sum-exp over V (single block; data hot in L2) ----------------
__global__ void lse_kernel(const float* __restrict__ logits, float* __restrict__ logZ)
{
    __shared__ float red[256];
    const int tid = threadIdx.x;
    float m = -INFINITY;
    for (int v = tid; v < VSZ; v += 256) m = fmaxf(m, logits[v]);
    red[tid] = m; __syncthreads();
    for (int s = 128; s > 0; s >>= 1) {
        if (tid < s) red[tid] = fmaxf(red[tid], red[tid + s]);
        __syncthreads();
    }
    const float gmax = red[0];
    __syncthreads();
    float s = 0.f;
    for (int v = tid; v < VSZ; v += 256) s += __expf(logits[v] - gmax);
    red[tid] = s; __syncthreads();
    for (int st = 128; st > 0; st >>= 1) {
        if (tid < st) red[tid] += red[tid + st];
        __syncthreads();
    }
    if (tid == 0) logZ[0] = gmax + __logf(red[0]);
}

__global__ void sub_kernel(float* __restrict__ d_out, const float* __restrict__ logZ)
{
    const int v = blockIdx.x * 256 + threadIdx.x;
    if (v < VSZ) d_out[v] -= logZ[0];
}

// ---------------- launch ----------------
extern "C" void kernel_launch(void* const* d_in, const int* in_sizes, int n_in,
                              void* d_out_v, int out_size, void* d_ws, size_t ws_size,
                              hipStream_t stream)
{
    const int*   tok    = (const int*)  d_in[0];
    const float* hidden = (const float*)d_in[1];
    const float* enc    = (const float*)d_in[2];
    const float* emb    = (const float*)d_in[3];
    const float* W_attn = (const float*)d_in[4];
    const float* b_attn = (const float*)d_in[5];
    const float* W_comb = (const float*)d_in[6];
    const float* b_comb = (const float*)d_in[7];
    const float* w_ih   = (const float*)d_in[8];
    const float* w_hh   = (const float*)d_in[9];
    const float* b_ih   = (const float*)d_in[10];
    const float* b_hh   = (const float*)d_in[11];
    const float* W_out  = (const float*)d_in[12];
    const float* b_out  = (const float*)d_in[13];
    float* d_out = (float*)d_out_v;
    float* ws    = (float*)d_ws;

    float* comb_in = ws;            // 2048
    float* xvec    = ws + 2048;     // 1024
    float* gi      = ws + 3072;     // 3072
    float* gh      = ws + 6144;     // 3072
    float* hnew    = ws + 9216;     // 1024
    float* logZ    = ws + 10240;    // 1

    attn_kernel<<<1, 256, 0, stream>>>(tok, hidden, enc, emb, W_attn, b_attn,
                                       d_out, comb_in);
    gemv_wmma<<<(HSZ + 127) / 128, 256, 0, stream>>>(W_comb, comb_in, b_comb,
                                                     xvec, HSZ, 2 * HSZ, 1);
    gemv_wmma<<<(3 * HSZ + 127) / 128, 256, 0, stream>>>(w_ih, xvec, b_ih,
                                                         gi, 3 * HSZ, HSZ, 0);
    gemv_wmma<<<(3 * HSZ + 127) / 128, 256, 0, stream>>>(w_hh, hidden, b_hh,
                                                         gh, 3 * HSZ, HSZ, 0);
    gru_kernel<<<(HSZ + 255) / 256, 256, 0, stream>>>(gi, gh, hidden, d_out, hnew);
    gemv_wmma<<<(VSZ + 127) / 128, 256, 0, stream>>>(W_out, hnew, b_out,
                                                     d_out, VSZ, HSZ, 0);
    lse_kernel<<<1, 256, 0, stream>>>(d_out, logZ);
    sub_kernel<<<(VSZ + 255) / 256, 256, 0, stream>>>(d_out, logZ);
}